// _ProposalTargetLayer_2310692405256
// MI455X (gfx1250) — compile-verified
//
#include <hip/hip_runtime.h>
#include <cstdint>

// ---------------------------------------------------------------------------
// Problem constants (from reference): B=16, N=16384, G=256, R=256, FG=64
// ---------------------------------------------------------------------------
#define BATCH        16
#define NROI         16384
#define NGT          256
#define NR_ALL       (NROI + NGT)        // 16640 rows of all_rois per image
#define RSAMP        256
#define FG_MAX       64
#define ROW_TILES    (NR_ALL / 16)       // 1040 row tiles of 16
#define WAVES_PER_BLK 8
#define BLKS_PER_IMG (ROW_TILES / WAVES_PER_BLK)  // 130

typedef float v2f __attribute__((ext_vector_type(2)));
typedef float v8f __attribute__((ext_vector_type(8)));
typedef unsigned int v4u __attribute__((ext_vector_type(4)));
typedef int  v4i __attribute__((ext_vector_type(4)));
typedef int  v8i __attribute__((ext_vector_type(8)));

// ---------------------------------------------------------------------------
// Tensor Data Mover: DMA a contiguous run of n_dwords DWORDs from global
// memory into LDS.  D# built per CDNA5 ISA.
// Arity differs between ROCm 7.2 (5 args) and amdgpu-toolchain (6 args);
// detect via the TDM header that only ships with the latter.
// ---------------------------------------------------------------------------
#if __has_builtin(__builtin_amdgcn_tensor_load_to_lds)
#define HAVE_TDM 1
#else
#define HAVE_TDM 0
#endif

// Compile-time probes (show up in stderr; do not affect codegen).
#if !HAVE_TDM
#warning "CDNA5 probe: __builtin_amdgcn_tensor_load_to_lds NOT available -> fallback LDS copy"
#elif __has_include(<hip/amd_detail/amd_gfx1250_TDM.h>)
#warning "CDNA5 probe: TDM builtin, 6-arg path (amdgpu-toolchain headers)"
#else
#warning "CDNA5 probe: TDM builtin, 5-arg path (ROCm 7.2)"
#endif
#if !__has_builtin(__builtin_amdgcn_s_wait_tensorcnt)
#warning "CDNA5 probe: s_wait_tensorcnt builtin NOT available -> inline asm"
#endif

__device__ __forceinline__ void tdm_load_1d_to_lds(const void* gsrc,
                                                   unsigned lds_byte_addr,
                                                   unsigned n_dwords) {
#if HAVE_TDM
  unsigned long long ga = (unsigned long long)(uintptr_t)gsrc;
  v4u g0;
  g0.x = 1u;                                    // count=1, user descriptor
  g0.y = lds_byte_addr;                         // lds_addr
  g0.z = (unsigned)(ga & 0xffffffffu);          // global_addr[31:0]
  g0.w = (unsigned)((ga >> 32) & 0x01ffffffu)   // global_addr[56:32]
       | (2u << 30);                            // type = 2 ("image")
  v8i g1;
  g1[0] = (int)(2u << 16);                      // data_size = 4 bytes
  g1[1] = (int)((n_dwords & 0xffffu) << 16);    // tensor_dim0[15:0]
  g1[2] = (int)((n_dwords >> 16) | (1u << 16)); // tensor_dim0[31:16] | tensor_dim1=1
  g1[3] = (int)((n_dwords & 0xffffu) << 16);    // tile_dim0 = n_dwords
  g1[4] = 1;                                    // tile_dim1 = 1
  g1[5] = (int)n_dwords;                        // tensor_dim0_stride[31:0]
  g1[6] = 0;
  g1[7] = 0;
  v4i gz = {0, 0, 0, 0};
#if __has_include(<hip/amd_detail/amd_gfx1250_TDM.h>)
  v8i gz8 = {0, 0, 0, 0, 0, 0, 0, 0};
  __builtin_amdgcn_tensor_load_to_lds(g0, g1, gz, gz, gz8, 0);
#else
  __builtin_amdgcn_tensor_load_to_lds(g0, g1, gz, gz, 0);
#endif
#else
  (void)gsrc; (void)lds_byte_addr; (void)n_dwords;
#endif
}

__device__ __forceinline__ void wait_tensorcnt0() {
#if __has_builtin(__builtin_amdgcn_s_wait_tensorcnt)
  __builtin_amdgcn_s_wait_tensorcnt(0);
#else
  asm volatile("s_wait_tensorcnt 0" ::: "memory");
#endif
}

__device__ __forceinline__ unsigned hashu(unsigned x) {
  x ^= x >> 16; x *= 0x7feb352du;
  x ^= x >> 15; x *= 0x846ca68bu;
  x ^= x >> 16;
  return x;
}

// ---------------------------------------------------------------------------
// Kernel 1: IoU of all_rois (=concat(rois, gt)) vs gt, max+argmax over gt.
// One wave per 16-row tile.  WMMA f32 16x16x4 computes the pairwise
// area_roi + area_gt outer-sum directly in the C/D layout; VALU computes the
// intersection in that same layout.  The running max keeps the IoU as a
// (numerator, denominator) pair so the hot loop needs no division at all:
//    inter/uni > bn/bu  <=>  inter*bu > bn*uni   (both denominators > 0)
// One real divide per output row at the very end.
// ---------------------------------------------------------------------------
__global__ __launch_bounds__(256) void iou_argmax_kernel(
    const float* __restrict__ rois, const float* __restrict__ gt,
    float* __restrict__ max_ov, int* __restrict__ gt_assign) {
  __shared__ __align__(16) float sh_gt[NGT * 4];            // 4 KB GT boxes
  __shared__ float sh_ag[NGT];                              // GT areas
  __shared__ float sh_roi[WAVES_PER_BLK][16][4];            // per-wave ROI tile

  const int b    = blockIdx.x / BLKS_PER_IMG;
  const int blk  = blockIdx.x % BLKS_PER_IMG;
  const int tid  = threadIdx.x;
  const int wave = tid >> 5;
  const int lane = tid & 31;
  const int halfLane = lane & 15;
  const bool lowHalf = lane < 16;

  // --- Stage this image's 256 GT boxes into LDS with the TDM --------------
#if HAVE_TDM
  if (tid == 0) {
    tdm_load_1d_to_lds(gt + (size_t)b * NGT * 4,
                       (unsigned)(uintptr_t)sh_gt, NGT * 4);
    wait_tensorcnt0();
  }
#else
  for (int i = tid; i < NGT * 4; i += 256)
    sh_gt[i] = gt[(size_t)b * NGT * 4 + i];
#endif
  __syncthreads();

  // Per-GT areas (256 threads == NGT).
  {
    float gx1 = sh_gt[tid * 4 + 0], gy1 = sh_gt[tid * 4 + 1];
    float gx2 = sh_gt[tid * 4 + 2], gy2 = sh_gt[tid * 4 + 3];
    sh_ag[tid] = (gx2 - gx1 + 1.f) * (gy2 - gy1 + 1.f);
  }
  __syncthreads();

  const int rowTile = blk * WAVES_PER_BLK + wave;
  const int baseRow = rowTile * 16;

  // Hint the next row tile into cache (global_prefetch_b8).
  if (baseRow + 16 < NROI)
    __builtin_prefetch(rois + ((size_t)b * NROI + baseRow + 16) * 4, 0, 1);

  // Lanes 0..15 stage the 16 ROI rows of this tile into LDS and compute the
  // A-fragment area (A layout: lanes 0-15 hold M=lane, K=0/1; lanes 16-31
  // hold K=2/3 which we zero-pad).
  float abA = 0.f;
  if (lowHalf) {
    const int row = baseRow + lane;
    const float* src = (row < NROI)
        ? (rois + ((size_t)b * NROI + row) * 4)
        : (gt   + ((size_t)b * NGT + (row - NROI)) * 4);
    float4 bx = *(const float4*)src;
    sh_roi[wave][lane][0] = bx.x;
    sh_roi[wave][lane][1] = bx.y;
    sh_roi[wave][lane][2] = bx.z;
    sh_roi[wave][lane][3] = bx.w;
    abA = (bx.z - bx.x + 1.f) * (bx.w - bx.y + 1.f);
  }

  // Pull the 8 rows this lane owns in the C/D layout into registers.
  float rx1[8], ry1[8], rx2[8], ry2[8];
#pragma unroll
  for (int v = 0; v < 8; ++v) {
    const int m = v + (lowHalf ? 0 : 8);
    rx1[v] = sh_roi[wave][m][0];
    ry1[v] = sh_roi[wave][m][1];
    rx2[v] = sh_roi[wave][m][2];
    ry2[v] = sh_roi[wave][m][3];
  }

  // A = [ab_i, 1, 0, 0] (16x4), constant across column tiles.
  v2f afrag;
  afrag.x = lowHalf ? abA : 0.f;   // K=0 (low) / K=2 (high, padded 0)
  afrag.y = lowHalf ? 1.f : 0.f;   // K=1 (low) / K=3 (high, padded 0)

  float bn[8], bu[8];
  int   bid[8];
#pragma unroll
  for (int v = 0; v < 8; ++v) { bn[v] = -1.f; bu[v] = 1.f; bid[v] = 0; }

  // 16 column tiles of 16 GT boxes each; unroll x2 so the WMMA of one tile
  // overlaps the VALU IoU of the previous one.
#pragma unroll 2
  for (int t = 0; t < 16; ++t) {
    const int n = t * 16 + halfLane;
    const float4 gq = *(const float4*)&sh_gt[n * 4];

    // B = [[1...1],[ag_j],[0],[0]] (4x16): VGPR0 = K0 (low half) / K2 (high),
    // VGPR1 = K1 / K3.
    v2f bfrag;
    bfrag.x = lowHalf ? 1.f : 0.f;
    bfrag.y = lowHalf ? sh_ag[n] : 0.f;

    v8f acc = {};
    // D[M,N] = ab[M] + ag[N]  (outer-sum part of the IoU denominator)
    acc = __builtin_amdgcn_wmma_f32_16x16x4_f32(
        false, afrag, false, bfrag, (short)0, acc, false, false);

#pragma unroll
    for (int v = 0; v < 8; ++v) {
      float ix1 = fmaxf(rx1[v], gq.x), iy1 = fmaxf(ry1[v], gq.y);
      float ix2 = fminf(rx2[v], gq.z), iy2 = fminf(ry2[v], gq.w);
      float iw = fmaxf(ix2 - ix1 + 1.f, 0.f);
      float ih = fmaxf(iy2 - iy1 + 1.f, 0.f);
      float inter = iw * ih;
      float uni = acc[v] - inter;           // ab + ag - inter, always >= 1
      // inter/uni > bn/bu  <=>  inter*bu > bn*uni  (no divide in hot loop)
      if (inter * bu[v] > bn[v] * uni) {
        bn[v] = inter; bu[v] = uni; bid[v] = n;
      }
    }
  }

  // Max+argmax across the 16-lane half (N axis) via fraction cross-multiply;
  // lowest index wins ties.  One real divide per output row at the end.
#pragma unroll
  for (int v = 0; v < 8; ++v) {
    float n0 = bn[v], d0 = bu[v];
    int i0 = bid[v];
#pragma unroll
    for (int off = 1; off < 16; off <<= 1) {
      float n1 = __shfl_xor(n0, off, 32);
      float d1 = __shfl_xor(d0, off, 32);
      int   i1 = __shfl_xor(i0, off, 32);
      float l = n1 * d0, r = n0 * d1;       // n1/d1 vs n0/d0
      if (l > r || (l == r && i1 < i0)) { n0 = n1; d0 = d1; i0 = i1; }
    }
    if (halfLane == v) {
      const int row = baseRow + v + (lowHalf ? 0 : 8);
      const size_t o = (size_t)b * NR_ALL + row;
      max_ov[o] = n0 / d0;
      gt_assign[o] = i0;
    }
  }
}

// ---------------------------------------------------------------------------
// Kernel 2: per-image fg/bg compaction (deterministic ballot+scan), hashed
// sampling (stands in for the jax threefry stream), gather + bbox_transform,
// and the four concatenated outputs.
// ---------------------------------------------------------------------------
__global__ __launch_bounds__(256) void sample_transform_kernel(
    const float* __restrict__ rois, const float* __restrict__ gt,
    const float* __restrict__ max_ov, const int* __restrict__ gt_assign,
    int* __restrict__ fgList, int* __restrict__ bgList,
    float* __restrict__ out) {
  const int b = blockIdx.x;
  const int tid = threadIdx.x;
  const int wave = tid >> 5;
  const int lane = tid & 31;

  __shared__ int sh_wfg[8], sh_wbg[8], sh_offF[8], sh_offB[8], sh_tot[2];
  if (tid == 0) { sh_tot[0] = 0; sh_tot[1] = 0; }
  __syncthreads();

  // Stream-compact fg / bg indices (deterministic: index order preserved).
  for (int base = 0; base < NR_ALL; base += 256) {
    const int i = base + tid;
    const float mv = max_ov[(size_t)b * NR_ALL + i];
    const bool fg = mv >= 0.5f;
    const bool bg = (mv < 0.5f) && (mv >= 0.1f);
    const unsigned mF = (unsigned)__ballot(fg);
    const unsigned mB = (unsigned)__ballot(bg);
    if (lane == 0) { sh_wfg[wave] = __popc(mF); sh_wbg[wave] = __popc(mB); }
    __syncthreads();
    if (tid == 0) {
      int aF = sh_tot[0], aB = sh_tot[1];
      for (int w = 0; w < 8; ++w) {
        const int tF = sh_wfg[w], tB = sh_wbg[w];
        sh_offF[w] = aF; sh_offB[w] = aB;
        aF += tF; aB += tB;
      }
      sh_tot[0] = aF; sh_tot[1] = aB;
    }
    __syncthreads();
    const unsigned lt = (1u << lane) - 1u;
    if (fg) fgList[(size_t)b * NR_ALL + sh_offF[wave] + __popc(mF & lt)] = i;
    if (bg) bgList[(size_t)b * NR_ALL + sh_offB[wave] + __popc(mB & lt)] = i;
    __syncthreads();
  }
  __threadfence();
  __syncthreads();
  const int nfg = sh_tot[0];
  const int nbg = sh_tot[1];

  // Sample R=256 slots (thread j owns slot j).
  const int j = tid;
  const int fg_count = (nbg > 0) ? min(FG_MAX, nfg) : ((nfg > 0) ? RSAMP : 0);
  const unsigned h1 = hashu(0x9E3779B9u * (unsigned)(b * RSAMP + j) + 0x85ebca6bu);
  const unsigned h2 = hashu(0xC2B2AE35u * (unsigned)(b * RSAMP + j) + 0x27d4eb2fu);
  int pick = 0;
  float label = 0.f;
  if (j < fg_count && nfg > 0) {
    pick = fgList[(size_t)b * NR_ALL + (int)(h1 % (unsigned)nfg)];
    label = 1.f;
  } else if (nbg > 0) {
    pick = bgList[(size_t)b * NR_ALL + (int)(h2 % (unsigned)nbg)];
  } else if (nfg > 0) {
    pick = fgList[(size_t)b * NR_ALL + (int)(h2 % (unsigned)nfg)];
  }

  // Gather + bbox_transform.
  const float* ep = (pick < NROI)
      ? rois + ((size_t)b * NROI + pick) * 4
      : gt   + ((size_t)b * NGT + (pick - NROI)) * 4;
  const float4 ex = *(const float4*)ep;
  const int ga = gt_assign[(size_t)b * NR_ALL + pick];
  const float4 gx = *(const float4*)(gt + ((size_t)b * NGT + ga) * 4);

  const float ew = ex.z - ex.x + 1.f, eh = ex.w - ex.y + 1.f;
  const float ecx = ex.x + 0.5f * ew, ecy = ex.y + 0.5f * eh;
  const float gw = gx.z - gx.x + 1.f, gh = gx.w - gx.y + 1.f;
  const float gcx = gx.x + 0.5f * gw, gcy = gx.y + 0.5f * gh;
  const float t0 = (gcx - ecx) / ew;
  const float t1 = (gcy - ecy) / eh;
  const float t2 = __logf(gw / ew);
  const float t3 = __logf(gh / eh);

  // Outputs concatenated flat: labels | rois_batch | bbox_targets | weights.
  const int s = b * RSAMP + j;
  out[s] = label;
  float* rb = out + (size_t)BATCH * RSAMP + (size_t)s * 4;
  rb[0] = ex.x; rb[1] = ex.y; rb[2] = ex.z; rb[3] = ex.w;
  float* bt = out + (size_t)BATCH * RSAMP * 5 + (size_t)s * 4;
  bt[0] = t0 * label; bt[1] = t1 * label; bt[2] = t2 * label; bt[3] = t3 * label;
  float* bw = out + (size_t)BATCH * RSAMP * 9 + (size_t)s * 4;
  bw[0] = label; bw[1] = label; bw[2] = label; bw[3] = label;
}

// ---------------------------------------------------------------------------
// Launch: workspace = max_ov | gt_assign | fgList | bgList (~4.1 MB).
// ---------------------------------------------------------------------------
extern "C" void kernel_launch(void* const* d_in, const int* in_sizes, int n_in,
                              void* d_out, int out_size, void* d_ws,
                              size_t ws_size, hipStream_t stream) {
  (void)in_sizes; (void)n_in; (void)out_size; (void)ws_size;
  const float* rois = (const float*)d_in[0];
  const float* gt   = (const float*)d_in[1];

  float* max_ov   = (float*)d_ws;
  int* gt_assign  = (int*)(max_ov + (size_t)BATCH * NR_ALL);
  int* fgList     = gt_assign + (size_t)BATCH * NR_ALL;
  int* bgList     = fgList + (size_t)BATCH * NR_ALL;

  iou_argmax_kernel<<<dim3(BATCH * BLKS_PER_IMG), 256, 0, stream>>>(
      rois, gt, max_ov, gt_assign);
  sample_transform_kernel<<<dim3(BATCH), 256, 0, stream>>>(
      rois, gt, max_ov, gt_assign, fgList, bgList, (float*)d_out);
}